// FeedForwardQuantum_65481071408245
// MI455X (gfx1250) — compile-verified
//
#include <hip/hip_runtime.h>

typedef float v2f __attribute__((ext_vector_type(2)));
typedef float v4f __attribute__((ext_vector_type(4)));
typedef float v8f __attribute__((ext_vector_type(8)));

namespace {
constexpr int kNQ     = 8;          // qubits = embed dim in/out
constexpr int kFFN    = 512;
constexpr int kTokens = 32 * 4096;  // B*S
constexpr int kWaves  = 8;          // waves per block (wave32)
constexpr int kMT     = 4;          // 16-token M-tiles per wave (64 tokens/wave)
}

// Fully fused, LDS-free: q = cos(x+phi) -> h^T = w1 q^T (+b1, ReLU) -> out^T = w2 h^T (+b2).
// GEMM1's D fragment is reused *directly* as GEMM2's B fragment via a K-permutation
// {2i,2i+1,2i+8,2i+9} per wmma step (w2 A-fragment loads use the same permutation).
// Biases are folded in as extra WMMA K-rows against a ones-column fragment.
__global__ __launch_bounds__(256)
void ffq_fused_wmma(const float* __restrict__ x,
                    const float* __restrict__ phi,
                    const float* __restrict__ w1,
                    const float* __restrict__ b1,
                    const float* __restrict__ w2,
                    const float* __restrict__ b2,
                    float* __restrict__ out)
{
  const int lane = threadIdx.x & 31;
  const int wave = threadIdx.x >> 5;
  const int row  = lane & 15;   // fragment M (A rows) or N (B/D cols)
  const int hi   = lane >> 4;   // half-wave: selects K sub-pair
  const int koff = hi << 1;     // K offset within a K=4 chunk

  const int tile0 = (blockIdx.x * kWaves + wave) * kMT;
  const float ph  = phi[0];

  // Quantum block collapses analytically: q[t][w] = cos(x[t][w] + phi).
  // Built directly as GEMM1 B-fragments (B(k,t) = q[t][k]); K=8 -> two K=4 chunks.
  v2f qlo[kMT], qhi[kMT];
#pragma unroll
  for (int m = 0; m < kMT; ++m) {
    const float* xr = x + (size_t)((tile0 + m) * 16 + row) * kNQ;
    qlo[m][0] = __cosf(xr[koff + 0] + ph);
    qlo[m][1] = __cosf(xr[koff + 1] + ph);
    qhi[m][0] = __cosf(xr[koff + 4] + ph);
    qhi[m][1] = __cosf(xr[koff + 5] + ph);
  }

  // Ones-column B fragment: virtual K-row of 1.0 used to inject biases via WMMA.
  v2f bone; bone[0] = hi ? 0.0f : 1.0f; bone[1] = 0.0f;
  const v8f vzero = {};

  // Seed out^T accumulators with b2 broadcast: D(m=e, n=t) += b2[e] * 1.
  // Rows e>=8 are address-clamped garbage that lands only in unstored lanes 16-31.
  v2f ab2; ab2[0] = hi ? 0.0f : b2[row & 7]; ab2[1] = 0.0f;
  const v8f accInit =
      __builtin_amdgcn_wmma_f32_16x16x4_f32(false, ab2, false, bone, (short)0, vzero, false, false);
  v8f acc[kMT];
#pragma unroll
  for (int m = 0; m < kMT; ++m) acc[m] = accInit;

  // w2 A-fragment base: w2[e][8*hi + ...], e clamped (pad rows never stored)
  const float* w2r = w2 + (size_t)(row & 7) * kFFN + 8 * hi;

  for (int n0 = 0; n0 < kFFN; n0 += 16) {
    // GEMM1 A-fragments: A(m,k) = w1[n0+m][k]
    const float* w1r = w1 + (size_t)(n0 + row) * kNQ;
    v2f wlo, whi;
    wlo[0] = w1r[koff + 0]; wlo[1] = w1r[koff + 1];
    whi[0] = w1r[koff + 4]; whi[1] = w1r[koff + 5];

    // b1 bias as WMMA: A(m, kvirt) = b1[n0+m] against ones column
    v2f ab1; ab1[0] = hi ? 0.0f : b1[n0 + row]; ab1[1] = 0.0f;

    // GEMM2 A-fragments, K-permuted: wa2[i] = w2[e][n0 + 2i + 8*hi + {0,1}]
    // -> 8 contiguous floats per lane, merges into two b128 loads.
    const v4f wq0 = *(const v4f*)(w2r + n0);
    const v4f wq1 = *(const v4f*)(w2r + n0 + 4);

    // Shared bias seed for all tiles' h accumulators (amortized over kMT)
    const v8f hinit =
        __builtin_amdgcn_wmma_f32_16x16x4_f32(false, ab1, false, bone, (short)0, vzero, false, false);

#pragma unroll
    for (int m = 0; m < kMT; ++m) {
      // h^T(16x16) = w1_chunk(16x8) x q^T(8x16) + b1
      v8f h = __builtin_amdgcn_wmma_f32_16x16x4_f32(false, wlo, false, qlo[m], (short)0, hinit, false, false);
      h     = __builtin_amdgcn_wmma_f32_16x16x4_f32(false, whi, false, qhi[m], (short)0, h,     false, false);
      // ReLU on D fragment: ternary form lowers to a single v_max_num_f32 per element
#pragma unroll
      for (int r = 0; r < 8; ++r) h[r] = h[r] > 0.0f ? h[r] : 0.0f;

      // GEMM2 partial: B fragment IS the (h[2i], h[2i+1]) register pair
      // (K order per step: {n0+2i, n0+2i+1, n0+2i+8, n0+2i+9})
#pragma unroll
      for (int i = 0; i < 4; ++i) {
        v2f a2, bfrag;
        a2[0]    = (i < 2) ? wq0[2 * i]     : wq1[2 * (i - 2)];
        a2[1]    = (i < 2) ? wq0[2 * i + 1] : wq1[2 * (i - 2) + 1];
        bfrag[0] = h[2 * i];
        bfrag[1] = h[2 * i + 1];
        acc[m] = __builtin_amdgcn_wmma_f32_16x16x4_f32(false, a2, false, bfrag, (short)0, acc[m], false, false);
      }
    }
  }

  // Epilogue: out^T D layout -> lanes 0-15 hold token row (e=0..7 contiguous in VGPRs 0..7)
  if (hi == 0) {
#pragma unroll
    for (int m = 0; m < kMT; ++m) {
      float* orow = out + (size_t)((tile0 + m) * 16 + row) * kNQ;
      v4f lo4, hi4;
      lo4[0] = acc[m][0]; lo4[1] = acc[m][1]; lo4[2] = acc[m][2]; lo4[3] = acc[m][3];
      hi4[0] = acc[m][4]; hi4[1] = acc[m][5]; hi4[2] = acc[m][6]; hi4[3] = acc[m][7];
      *(v4f*)(orow + 0) = lo4;   // two global_store_b128 per token
      *(v4f*)(orow + 4) = hi4;
    }
  }
}

extern "C" void kernel_launch(void* const* d_in, const int* in_sizes, int n_in,
                              void* d_out, int out_size, void* d_ws, size_t ws_size,
                              hipStream_t stream) {
  const float* x   = (const float*)d_in[0];
  const float* phi = (const float*)d_in[1];
  const float* w1  = (const float*)d_in[2];
  const float* b1  = (const float*)d_in[3];
  const float* w2  = (const float*)d_in[4];
  const float* b2  = (const float*)d_in[5];
  float* out = (float*)d_out;

  const int blocks = kTokens / (16 * kMT * kWaves);   // 256 blocks of 8 waves
  ffq_fused_wmma<<<blocks, 256, 0, stream>>>(x, phi, w1, b1, w2, b2, out);
}